// GCNRegressor_76441827934616
// MI455X (gfx1250) — compile-verified
//
#include <hip/hip_runtime.h>

typedef __attribute__((ext_vector_type(2))) float v2f;
typedef __attribute__((ext_vector_type(8))) float v8f;

#define NN   50000
#define NE   1600000
#define DIM  64
#define EPSV 1e-5f

// ---------------- degree / normalization ----------------

__global__ void k_init_deg(float* __restrict__ deg) {
    int i = blockIdx.x * blockDim.x + threadIdx.x;
    if (i < NN) deg[i] = 1.0f;  // self-loop contribution
}

__global__ void k_deg_edges(const int* __restrict__ dst, float* __restrict__ deg) {
    int e = blockIdx.x * blockDim.x + threadIdx.x;
    if (e < NE) atomicAdd(&deg[dst[e]], 1.0f);
}

__global__ void k_dinv(float* __restrict__ deg) {
    int i = blockIdx.x * blockDim.x + threadIdx.x;
    if (i < NN) {
        float d = deg[i];
        deg[i] = (d > 0.0f) ? rsqrtf(d) : 0.0f;  // becomes dinv in place
    }
}

// ---------------- dense GEMM via f32 WMMA ----------------
// One wave computes a 16-row x 64-col tile of T = X @ W.
// A (16x4 f32): lanes 0-15 hold M=lane, K={k0,k0+1}; lanes 16-31 hold K={k0+2,k0+3}.
// B (4x16 f32): mirrored K split, N = lane&15.
// C/D (16x16 f32): VGPR r holds M = (lane>=16 ? 8 : 0) + r, N = lane&15.
__global__ __launch_bounds__(256) void k_gemm_wmma(const float* __restrict__ X,
                                                   const float* __restrict__ W,
                                                   float* __restrict__ T) {
    int wave = blockIdx.x * (blockDim.x >> 5) + (threadIdx.x >> 5);
    int row0 = wave * 16;
    if (row0 >= NN) return;               // uniform per wave: EXEC stays all-ones
    int lane = threadIdx.x & 31;
    int half = lane >> 4;                 // 0 or 1
    int lm   = lane & 15;

    v8f acc0 = {}, acc1 = {}, acc2 = {}, acc3 = {};
    const float* xr = X + (size_t)(row0 + lm) * DIM;

#pragma unroll
    for (int ks = 0; ks < 16; ++ks) {
        int kb = ks * 4 + half * 2;
        v2f a;
        a.x = xr[kb];
        a.y = xr[kb + 1];
        const float* w0 = W + (size_t)kb * DIM;
        v2f b0; b0.x = w0[lm];        b0.y = w0[DIM + lm];
        v2f b1; b1.x = w0[16 + lm];   b1.y = w0[DIM + 16 + lm];
        v2f b2; b2.x = w0[32 + lm];   b2.y = w0[DIM + 32 + lm];
        v2f b3; b3.x = w0[48 + lm];   b3.y = w0[DIM + 48 + lm];
        acc0 = __builtin_amdgcn_wmma_f32_16x16x4_f32(false, a, false, b0, (short)0, acc0, false, false);
        acc1 = __builtin_amdgcn_wmma_f32_16x16x4_f32(false, a, false, b1, (short)0, acc1, false, false);
        acc2 = __builtin_amdgcn_wmma_f32_16x16x4_f32(false, a, false, b2, (short)0, acc2, false, false);
        acc3 = __builtin_amdgcn_wmma_f32_16x16x4_f32(false, a, false, b3, (short)0, acc3, false, false);
    }

    float* trow = T + (size_t)(row0 + half * 8) * DIM;
#pragma unroll
    for (int r = 0; r < 8; ++r) {
        float* o = trow + (size_t)r * DIM;
        o[lm]      = acc0[r];
        o[16 + lm] = acc1[r];
        o[32 + lm] = acc2[r];
        o[48 + lm] = acc3[r];
    }
}

// ---------------- aggregation (self-loop + bias init, then edge scatter) ----------------

__global__ void k_agg_init(const float* __restrict__ T, const float* __restrict__ dinv,
                           const float* __restrict__ bias, float* __restrict__ O) {
    int g = blockIdx.x * blockDim.x + threadIdx.x;   // one float4 per thread
    if (g >= NN * (DIM / 4)) return;
    int node = g >> 4;
    int c4   = (g & 15) * 4;
    float d  = dinv[node];
    float s  = d * d;                                 // self-loop norm
    const float4 t = *reinterpret_cast<const float4*>(T + (size_t)node * DIM + c4);
    float4 o;
    o.x = t.x * s + bias[c4 + 0];
    o.y = t.y * s + bias[c4 + 1];
    o.z = t.z * s + bias[c4 + 2];
    o.w = t.w * s + bias[c4 + 3];
    *reinterpret_cast<float4*>(O + (size_t)node * DIM + c4) = o;
}

// 16 lanes cooperate per edge: coalesced 256B row gather, 4 L2 atomic adds per lane.
__global__ void k_agg_edges(const float* __restrict__ T, const int* __restrict__ src,
                            const int* __restrict__ dst, const float* __restrict__ dinv,
                            float* __restrict__ O) {
    unsigned int tid = blockIdx.x * blockDim.x + threadIdx.x;
    unsigned int e   = tid >> 4;
    if (e >= NE) return;
    int l = (int)(tid & 15);
    int s = src[e], d = dst[e];
    float w = dinv[s] * dinv[d];
    const float4 v = *reinterpret_cast<const float4*>(T + (size_t)s * DIM + l * 4);
    float* o = O + (size_t)d * DIM + l * 4;
    atomicAdd(o + 0, v.x * w);
    atomicAdd(o + 1, v.y * w);
    atomicAdd(o + 2, v.z * w);
    atomicAdd(o + 3, v.w * w);
}

// ---------------- batch norm (biased variance, as reference) ----------------

__global__ __launch_bounds__(256) void k_bn_stats(const float* __restrict__ O,
                                                  float* __restrict__ stats) {
    __shared__ float ssum[256];
    __shared__ float ssq[256];
    int f = blockIdx.x;                 // one block per feature column
    float s = 0.0f, q = 0.0f;
    for (int r = threadIdx.x; r < NN; r += 256) {
        float v = O[(size_t)r * DIM + f];
        s += v; q += v * v;
    }
    ssum[threadIdx.x] = s; ssq[threadIdx.x] = q;
    __syncthreads();
    for (int st = 128; st > 0; st >>= 1) {
        if (threadIdx.x < st) {
            ssum[threadIdx.x] += ssum[threadIdx.x + st];
            ssq[threadIdx.x]  += ssq[threadIdx.x + st];
        }
        __syncthreads();
    }
    if (threadIdx.x == 0) {
        float mu  = ssum[0] / (float)NN;
        float var = ssq[0] / (float)NN - mu * mu;
        stats[f]       = mu;
        stats[DIM + f] = var;
    }
}

__global__ void k_bn_relu(const float* __restrict__ O, const float* __restrict__ stats,
                          const float* __restrict__ g, const float* __restrict__ bt,
                          float* __restrict__ H) {
    int i = blockIdx.x * blockDim.x + threadIdx.x;
    if (i >= NN * DIM) return;
    int f = i & (DIM - 1);
    float mu = stats[f], var = stats[DIM + f];
    float y = (O[i] - mu) * rsqrtf(var + EPSV) * g[f] + bt[f];
    H[i] = fmaxf(y, 0.0f);
}

// ---------------- layer 3: matvec + scalar aggregation ----------------

__global__ void k_mv(const float* __restrict__ H, const float* __restrict__ W3,
                     float* __restrict__ t3) {
    int i = blockIdx.x * blockDim.x + threadIdx.x;
    if (i >= NN) return;
    const float4* h = reinterpret_cast<const float4*>(H + (size_t)i * DIM);
    const float4* w = reinterpret_cast<const float4*>(W3);
    float s = 0.0f;
#pragma unroll
    for (int j = 0; j < DIM / 4; ++j) {
        float4 a = h[j], b = w[j];
        s += a.x * b.x + a.y * b.y + a.z * b.z + a.w * b.w;
    }
    t3[i] = s;
}

__global__ void k_out_init(const float* __restrict__ t3, const float* __restrict__ dinv,
                           const float* __restrict__ b3, float* __restrict__ out) {
    int i = blockIdx.x * blockDim.x + threadIdx.x;
    if (i >= NN) return;
    float d = dinv[i];
    out[i] = t3[i] * d * d + b3[0];
}

__global__ void k_out_edges(const float* __restrict__ t3, const int* __restrict__ src,
                            const int* __restrict__ dst, const float* __restrict__ dinv,
                            float* __restrict__ out) {
    int e = blockIdx.x * blockDim.x + threadIdx.x;
    if (e >= NE) return;
    int s = src[e], d = dst[e];
    atomicAdd(&out[d], t3[s] * dinv[s] * dinv[d]);
}

// ---------------- launch ----------------

extern "C" void kernel_launch(void* const* d_in, const int* in_sizes, int n_in,
                              void* d_out, int out_size, void* d_ws, size_t ws_size,
                              hipStream_t stream) {
    (void)in_sizes; (void)n_in; (void)out_size; (void)ws_size;

    const float* x   = (const float*)d_in[0];
    const float* W1  = (const float*)d_in[1];
    const float* b1  = (const float*)d_in[2];
    const float* g1  = (const float*)d_in[3];
    const float* bt1 = (const float*)d_in[4];
    const float* W2  = (const float*)d_in[5];
    const float* b2  = (const float*)d_in[6];
    const float* g2  = (const float*)d_in[7];
    const float* bt2 = (const float*)d_in[8];
    const float* W3  = (const float*)d_in[9];
    const float* b3  = (const float*)d_in[10];
    const int*   ei  = (const int*)d_in[11];
    const int* src = ei;
    const int* dst = ei + NE;
    float* out = (float*)d_out;

    // workspace layout (256B aligned slices)
    char* ws = (char*)d_ws;
    float* dinv  = (float*)(ws + 0);                        // 200,000 B
    float* bufT  = (float*)(ws + 262144);                   // 12,800,000 B
    float* bufO  = (float*)(ws + 262144 + 12800000);        // 12,800,000 B
    float* t3    = (float*)(ws + 262144 + 2 * 12800000);    // 200,000 B
    float* stats = (float*)(ws + 262144 + 2 * 12800000 + 262144);  // 512 B

    const int TPB = 256;
    const int gemmWaves  = NN / 16;                         // 3125 tiles of 16 rows
    const int gemmBlocks = (gemmWaves + 7) / 8;             // 8 waves / block
    const int nodeBlocks  = (NN + TPB - 1) / TPB;
    const int edgeBlocks  = (NE + TPB - 1) / TPB;
    const int elemBlocks  = (NN * DIM + TPB - 1) / TPB;
    const int vec4Blocks  = (NN * (DIM / 4) + TPB - 1) / TPB;
    const int edge16Blocks = (int)(((unsigned)NE * 16u + TPB - 1) / TPB);

    // normalization coefficients
    k_init_deg <<<nodeBlocks, TPB, 0, stream>>>(dinv);
    k_deg_edges<<<edgeBlocks, TPB, 0, stream>>>(dst, dinv);
    k_dinv     <<<nodeBlocks, TPB, 0, stream>>>(dinv);

    // ---- layer 1: T1 = x@W1 (WMMA); O1 = aggregate + b1; H1 = relu(BN(O1)) -> bufT
    k_gemm_wmma<<<gemmBlocks, TPB, 0, stream>>>(x, W1, bufT);
    k_agg_init <<<vec4Blocks, TPB, 0, stream>>>(bufT, dinv, b1, bufO);
    k_agg_edges<<<edge16Blocks, TPB, 0, stream>>>(bufT, src, dst, dinv, bufO);
    k_bn_stats <<<DIM, TPB, 0, stream>>>(bufO, stats);
    k_bn_relu  <<<elemBlocks, TPB, 0, stream>>>(bufO, stats, g1, bt1, bufT);

    // ---- layer 2: ping-pong buffers; H2 -> bufO
    k_gemm_wmma<<<gemmBlocks, TPB, 0, stream>>>(bufT, W2, bufO);
    k_agg_init <<<vec4Blocks, TPB, 0, stream>>>(bufO, dinv, b2, bufT);
    k_agg_edges<<<edge16Blocks, TPB, 0, stream>>>(bufO, src, dst, dinv, bufT);
    k_bn_stats <<<DIM, TPB, 0, stream>>>(bufT, stats);
    k_bn_relu  <<<elemBlocks, TPB, 0, stream>>>(bufT, stats, g2, bt2, bufO);

    // ---- layer 3: scalar head
    k_mv       <<<nodeBlocks, TPB, 0, stream>>>(bufO, W3, t3);
    k_out_init <<<nodeBlocks, TPB, 0, stream>>>(t3, dinv, b3, out);
    k_out_edges<<<edgeBlocks, TPB, 0, stream>>>(t3, src, dst, dinv, out);
}